// FSattack_60627758350707
// MI455X (gfx1250) — compile-verified
//
#include <hip/hip_runtime.h>

// Problem sizes
constexpr int NN = 8192;   // nodes
constexpr int FF = 512;    // feature dim
constexpr float EPS = 1e-5f;

typedef float v2f __attribute__((ext_vector_type(2)));
typedef float v8f __attribute__((ext_vector_type(8)));

// ---------------------------------------------------------------------------
// Async copy: global (saddr base + 32-bit vaddr byte offset) -> LDS, 16 bytes.
// vdst operand is the wave-relative LDS byte offset == low 32 bits of the
// flat pointer (LDS aperture lives entirely in addr[63:32] per CDNA5 ISA).
// Tracked by ASYNCcnt; retires in issue order.
// ---------------------------------------------------------------------------
__device__ __forceinline__ void async_b128(unsigned lds_off, unsigned voff,
                                           const float* base) {
  asm volatile("global_load_async_to_lds_b128 %0, %1, %2"
               :: "v"(lds_off), "v"(voff), "s"(base) : "memory");
}

// ---------------------------------------------------------------------------
// Pass 1: one sweep over adj producing rowsum (direct) and per-block column
// partials (ws), read-coalesced. 64 blocks x 256 threads, 128 rows per block.
// ---------------------------------------------------------------------------
__global__ __launch_bounds__(256) void k_rowcol(const float* __restrict__ adj,
                                                float* __restrict__ rowsum,
                                                float* __restrict__ colpart) {
  const int tid = threadIdx.x;
  const int b = blockIdx.x;
  const int lane = tid & 31, w = tid >> 5;
  __shared__ float wsum[8];
  float colAcc[32];
#pragma unroll
  for (int k = 0; k < 32; ++k) colAcc[k] = 0.f;

  for (int rr = 0; rr < 128; ++rr) {
    const size_t r = (size_t)b * 128 + rr;
    const float* __restrict__ row = adj + r * NN;
    float rs = 0.f;
#pragma unroll
    for (int k = 0; k < 32; ++k) {
      float v = row[tid + (k << 8)];
      rs += v;
      colAcc[k] += v;
    }
#pragma unroll
    for (int off = 16; off > 0; off >>= 1) rs += __shfl_down(rs, off);
    if (lane == 0) wsum[w] = rs;
    __syncthreads();
    if (tid == 0) {
      float s = 0.f;
#pragma unroll
      for (int j = 0; j < 8; ++j) s += wsum[j];
      rowsum[r] = s;
    }
    __syncthreads();
  }
#pragma unroll
  for (int k = 0; k < 32; ++k)
    colpart[(size_t)b * NN + tid + (k << 8)] = colAcc[k];
}

// ---------------------------------------------------------------------------
// Pass 2: reduce column partials (fixed order -> deterministic), compute
// d = 0.5*(rowsum+colsum), z = (d+eps)^-1/2, w1 = z^2*d.
// ---------------------------------------------------------------------------
__global__ __launch_bounds__(256) void k_degree(const float* __restrict__ rowsum,
                                                const float* __restrict__ colpart,
                                                float* __restrict__ z,
                                                float* __restrict__ w1) {
  const int c = blockIdx.x * 256 + threadIdx.x;
  float s = 0.f;
  for (int blk = 0; blk < 64; ++blk) s += colpart[(size_t)blk * NN + c];
  float d = 0.5f * (rowsum[c] + s);
  float zi = rsqrtf(d + EPS);
  if (!isfinite(zi)) zi = 0.f;  // mirror reference's isinf guard
  z[c] = zi;
  w1[c] = zi * zi * d;
}

// ---------------------------------------------------------------------------
// Xsp = diag(z)*X stored in the GEMM's paired-K tiled layout:
//   Xsp[((ft*4096 + k/2)*128 + f%128)*2 + (k&1)],  ft = f/128
// i.e. per f-tile, K-pairs of a column are adjacent (float2). A 16-byte async
// load from Xsp then lands in LDS exactly in B-fragment order.
// Each thread: reads float2 from rows 2k2 and 2k2+1, writes one float4.
// ---------------------------------------------------------------------------
__global__ __launch_bounds__(256) void k_xscale(const float* __restrict__ X,
                                                const float* __restrict__ z,
                                                float* __restrict__ Xsp) {
  const int idx = blockIdx.x * 256 + threadIdx.x;   // [0, 4096*256)
  const int k2 = idx >> 8;                          // K-pair index, 0..4095
  const int fp = (idx & 255) << 1;                  // even feature col, 0..510
  const int ft = fp >> 7;                           // f-tile (128 wide)
  const int fm = fp & 127;
  const float z0 = z[2 * k2], z1 = z[2 * k2 + 1];
  float2 x0 = *(const float2*)(X + (size_t)(2 * k2) * FF + fp);
  float2 x1 = *(const float2*)(X + (size_t)(2 * k2 + 1) * FF + fp);
  float4 o;
  o.x = x0.x * z0;  // (k even, fm)
  o.y = x1.x * z1;  // (k odd,  fm)
  o.z = x0.y * z0;  // (k even, fm+1)
  o.w = x1.y * z1;  // (k odd,  fm+1)
  *(float4*)(Xsp + (((size_t)ft * 4096 + k2) * 128 + fm) * 2) = o;
}

// ---------------------------------------------------------------------------
// term1 = sum_i w1[i] * ||X_i||^2. 64 blocks, each wave handles 16 rows.
// ---------------------------------------------------------------------------
__global__ __launch_bounds__(256) void k_term1(const float* __restrict__ X,
                                               const float* __restrict__ w1,
                                               float* __restrict__ t1part) {
  const int tid = threadIdx.x, lane = tid & 31, w = tid >> 5;
  const int b = blockIdx.x;
  __shared__ float wacc[8];
  float local = 0.f;
  for (int j = 0; j < 16; ++j) {
    const int r = b * 128 + w * 16 + j;
    const float* __restrict__ xr = X + (size_t)r * FF;
    float s = 0.f;
#pragma unroll
    for (int k = 0; k < 16; ++k) {
      float v = xr[lane + (k << 5)];
      s += v * v;
    }
#pragma unroll
    for (int off = 16; off > 0; off >>= 1) s += __shfl_down(s, off);
    if (lane == 0) local += w1[r] * s;
  }
  if (lane == 0) wacc[w] = local;
  __syncthreads();
  if (tid == 0) {
    float s = 0.f;
#pragma unroll
    for (int j = 0; j < 8; ++j) s += wacc[j];
    t1part[b] = s;
  }
}

// ---------------------------------------------------------------------------
// term2 = sum(Xs . (adj @ Xs)) via fused fp32 WMMA GEMM-dot.
// Block tile 64 (M) x 128 (N), K = 8192 in chunks of 16.
// Double-buffered async staging (GLOBAL_LOAD_ASYNC_TO_LDS_B128, ASYNCcnt):
//   issue next tile (3 x b128/thread) -> s_wait_asynccnt 3 -> barrier ->
//   16 x v_wmma_f32_16x16x4_f32 -> barrier.
// A tile: row-major stride 20 (fragment = aligned b64, bank-conflict-free).
// B tile: paired-K float2 layout, k2-row stride 144 float2 (288 floats):
//   fragment = one aligned b64; lanes 0-15 (k2) and 16-31 (k2+1) hit
//   disjoint bank halves (288 mod 64 == 32).
// ---------------------------------------------------------------------------
__global__ __launch_bounds__(256) void k_term2(const float* __restrict__ adj,
                                               const float* __restrict__ Xsp,
                                               float* __restrict__ t2part) {
  __shared__ __align__(16) float Alds[2][64 * 20];    // adj tile 64x16
  __shared__ __align__(16) float Blds[2][8 * 288];    // Xs tile 16x128 paired
  __shared__ float red[256];

  const int tid = threadIdx.x;
  const int lane = tid & 31, w = tid >> 5;
  const int ml = lane & 15, half = lane >> 4;
  const int wm = w & 3, wn = w >> 2;
  const int i0 = blockIdx.x * 64;
  const int ft = blockIdx.y;          // f-tile index, f0 = ft*128
  constexpr int NK = NN / 16;

  // per-thread staging slots
  const int arow = tid >> 2, ac4 = (tid & 3) << 2;     // A: 1 x b128
  // B: 2 x b128; idx -> (k2row 0..7, even col n2 0..126)
  const int b0k2 = tid >> 6,        b0n2 = (tid & 63) << 1;
  const int b1k2 = (tid + 256) >> 6, b1n2 = b0n2;

  unsigned ldsA[2], ldsB0[2], ldsB1[2];
#pragma unroll
  for (int bb = 0; bb < 2; ++bb) {
    ldsA[bb]  = (unsigned)(size_t)&Alds[bb][arow * 20 + ac4];
    ldsB0[bb] = (unsigned)(size_t)&Blds[bb][b0k2 * 288 + 2 * b0n2];
    ldsB1[bb] = (unsigned)(size_t)&Blds[bb][b1k2 * 288 + 2 * b1n2];
  }
  unsigned voffA  = (unsigned)(((i0 + arow) * NN + ac4) * 4);          // k0=0
  const unsigned ftbase = (unsigned)ft * 4096u * 128u * 2u * 4u;       // bytes
  unsigned voffB0 = ftbase + (unsigned)((b0k2 * 128 + b0n2) * 2 * 4);
  unsigned voffB1 = ftbase + (unsigned)((b1k2 * 128 + b1n2) * 2 * 4);

  v8f acc[4];
#pragma unroll
  for (int ns = 0; ns < 4; ++ns)
#pragma unroll
    for (int v = 0; v < 8; ++v) acc[ns][v] = 0.f;

  // prologue: fill buffer 0
  async_b128(ldsA[0], voffA, adj);
  async_b128(ldsB0[0], voffB0, Xsp);
  async_b128(ldsB1[0], voffB1, Xsp);
  voffA += 16 * 4;            // K advances 16 floats
  voffB0 += 8 * 128 * 2 * 4;  // 8 k2-rows
  voffB1 += 8 * 128 * 2 * 4;

  for (int kb = 0; kb < NK; ++kb) {
    const int cur = kb & 1;
    if (kb + 1 < NK) {
      const int nxt = cur ^ 1;
      async_b128(ldsA[nxt], voffA, adj);
      async_b128(ldsB0[nxt], voffB0, Xsp);
      async_b128(ldsB1[nxt], voffB1, Xsp);
      voffA += 16 * 4;
      voffB0 += 8 * 128 * 2 * 4;
      voffB1 += 8 * 128 * 2 * 4;
      asm volatile("s_wait_asynccnt 0x3" ::: "memory");  // prev group done
    } else {
      asm volatile("s_wait_asynccnt 0x0" ::: "memory");  // last tile done
    }
    __syncthreads();

    const float* __restrict__ A = Alds[cur];
    const float* __restrict__ B = Blds[cur];
#pragma unroll
    for (int kk = 0; kk < 16; kk += 4) {
      // A frag (16x4 fp32): lanes 0-15 -> K = kk,kk+1 ; lanes 16-31 -> kk+2,kk+3
      v2f a = *(const v2f*)&A[(wm * 16 + ml) * 20 + kk + 2 * half];
#pragma unroll
      for (int ns = 0; ns < 4; ++ns) {
        const int n = wn * 64 + ns * 16 + ml;
        // paired layout: one aligned b64 gives {B[kk+2h][n], B[kk+2h+1][n]}
        v2f bf = *(const v2f*)&B[((kk >> 1) + half) * 288 + 2 * n];
        acc[ns] = __builtin_amdgcn_wmma_f32_16x16x4_f32(
            /*neg_a=*/false, a, /*neg_b=*/false, bf,
            /*c_mod=*/(short)0, acc[ns], /*reuse_a=*/false, /*reuse_b=*/false);
      }
    }
    __syncthreads();  // protects spare buffer from next iteration's async writes
  }

  // epilogue: partial += C[m,n] * Xs[i,n] with Xsp paired indexing.
  // C/D layout: VGPR v, lanes 0-15 -> M=v, lanes 16-31 -> M=v+8, N = lane%16.
  float partial = 0.f;
#pragma unroll
  for (int ns = 0; ns < 4; ++ns) {
    const int fm = wn * 64 + ns * 16 + ml;   // 0..127 within the f-tile
#pragma unroll
    for (int v = 0; v < 8; ++v) {
      const int i = i0 + wm * 16 + v + 8 * half;
      const size_t off = (((size_t)ft * 4096 + (i >> 1)) * 128 + fm) * 2 + (i & 1);
      partial += acc[ns][v] * Xsp[off];
    }
  }
  red[tid] = partial;
  __syncthreads();
#pragma unroll
  for (int s = 128; s > 0; s >>= 1) {
    if (tid < s) red[tid] += red[tid + s];
    __syncthreads();
  }
  if (tid == 0) t2part[blockIdx.y * 128 + blockIdx.x] = red[0];
}

// ---------------------------------------------------------------------------
// Final: out = sum(t1part) - sum(t2part), fixed-order tree reduction.
// ---------------------------------------------------------------------------
__global__ __launch_bounds__(256) void k_final(const float* __restrict__ t1part,
                                               const float* __restrict__ t2part,
                                               float* __restrict__ out) {
  __shared__ float red[256];
  const int tid = threadIdx.x;
  float s = 0.f;
  if (tid < 64) s += t1part[tid];
  s -= t2part[tid];
  s -= t2part[tid + 256];
  red[tid] = s;
  __syncthreads();
#pragma unroll
  for (int st = 128; st > 0; st >>= 1) {
    if (tid < st) red[tid] += red[tid + st];
    __syncthreads();
  }
  if (tid == 0) out[0] = red[0];
}

// ---------------------------------------------------------------------------
extern "C" void kernel_launch(void* const* d_in, const int* in_sizes, int n_in,
                              void* d_out, int out_size, void* d_ws, size_t ws_size,
                              hipStream_t stream) {
  (void)in_sizes; (void)n_in; (void)out_size; (void)ws_size;
  const float* adj = (const float*)d_in[0];  // [8192, 8192] f32
  const float* X   = (const float*)d_in[1];  // [8192, 512]  f32
  float* out = (float*)d_out;                // scalar
  float* ws  = (float*)d_ws;

  // ws layout (floats): Xsp[8192*512] | z[8192] | w1[8192] | rowsum[8192]
  //                     | colpart[64*8192] | t1part[64] | t2part[512]
  // total ~= 18.1 MB of scratch
  float* Xsp     = ws;
  float* z       = Xsp + (size_t)NN * FF;
  float* w1      = z + NN;
  float* rowsum  = w1 + NN;
  float* colpart = rowsum + NN;
  float* t1part  = colpart + 64 * (size_t)NN;
  float* t2part  = t1part + 64;

  k_rowcol<<<64, 256, 0, stream>>>(adj, rowsum, colpart);
  k_degree<<<32, 256, 0, stream>>>(rowsum, colpart, z, w1);
  k_xscale<<<(NN / 2) * (FF / 2) / 256, 256, 0, stream>>>(X, z, Xsp);
  k_term1 <<<64, 256, 0, stream>>>(X, w1, t1part);
  k_term2 <<<dim3(128, 4), 256, 0, stream>>>(adj, Xsp, t2part);
  k_final <<<1, 256, 0, stream>>>(t1part, t2part, out);
}